// GatedGCNLSPELayer_1168231104593
// MI455X (gfx1250) — compile-verified
//
#include <hip/hip_runtime.h>
#include <math.h>

typedef float v2f __attribute__((ext_vector_type(2)));
typedef float v8f __attribute__((ext_vector_type(8)));

#define DD 64
#define LDS_STRIDE 68
#define EPS_ETA 1e-6f
#define EPS_BN 1e-5f

// ---------------- WMMA helpers (V_WMMA_F32_16X16X4_F32) ----------------
// A: 16x4 (MxK) f32, 2 VGPRs: lane<16 -> row=lane, v0=K0,v1=K1; lane>=16 -> row=lane-16, v0=K2,v1=K3
// B: 4x16 (KxN) f32, mirrored with N instead of M.
// C/D: 16x16 f32, 8 VGPRs: vj, lane<16 -> (M=j, N=lane); lane>=16 -> (M=j+8, N=lane-16)

__device__ __forceinline__ v8f bias_frag(const float* __restrict__ b, int col_base) {
  const int lane = threadIdx.x & 31;
  float v = b[col_base + (lane & 15)];
  v8f acc;
#pragma unroll
  for (int j = 0; j < 8; ++j) acc[j] = v;
  return acc;
}

__device__ __forceinline__ v8f wmma_k64(const float* At /* LDS 16 x LDS_STRIDE */,
                                        const float* __restrict__ W /* 64x64 row-major [k][n] */,
                                        int col_base, v8f acc) {
  const int lane = threadIdx.x & 31;
  const int half = lane >> 4;
  const int r = lane & 15;
#pragma unroll
  for (int k0 = 0; k0 < 64; k0 += 4) {
    v2f a, b;
    a.x = At[r * LDS_STRIDE + k0 + 2 * half + 0];
    a.y = At[r * LDS_STRIDE + k0 + 2 * half + 1];
    b.x = W[(k0 + 2 * half + 0) * DD + col_base + r];
    b.y = W[(k0 + 2 * half + 1) * DD + col_base + r];
    acc = __builtin_amdgcn_wmma_f32_16x16x4_f32(false, a, false, b, (short)0, acc, false, false);
  }
  return acc;
}

__device__ __forceinline__ void store_tile(float* __restrict__ out, int row_base,
                                           int col_base, v8f acc, int nrows) {
  const int lane = threadIdx.x & 31;
  const int half = lane >> 4;
  const int r = lane & 15;
#pragma unroll
  for (int j = 0; j < 8; ++j) {
    int row = row_base + j + 8 * half;
    if (row < nrows) out[(size_t)row * DD + col_base + r] = acc[j];
  }
}

// ---------------- K0: zero scratch ----------------
__global__ void zero_kernel(float* p, size_t n) {
  size_t i = (size_t)blockIdx.x * blockDim.x + threadIdx.x;
  size_t stride = (size_t)gridDim.x * blockDim.x;
  for (; i < n; i += stride) p[i] = 0.f;
}

// ---------------- K1: fused node GEMMs ----------------
__global__ __launch_bounds__(128) void node_linear_kernel(
    const float* __restrict__ h, const float* __restrict__ p,
    const float* __restrict__ A1w, const float* __restrict__ A1b,
    const float* __restrict__ A2w, const float* __restrict__ A2b,
    const float* __restrict__ B1w, const float* __restrict__ B1b,
    const float* __restrict__ B2w, const float* __restrict__ B2b,
    const float* __restrict__ C1w, const float* __restrict__ C1b,
    const float* __restrict__ C2w, const float* __restrict__ C2b,
    float* __restrict__ A1h, float* __restrict__ A2hp,
    float* __restrict__ B1h, float* __restrict__ B2h,
    float* __restrict__ C1p, float* __restrict__ C2p, int n) {
  __shared__ float ht[16 * LDS_STRIDE];
  __shared__ float pt[16 * LDS_STRIDE];
  const int row_base = blockIdx.x * 16;
  for (int i = threadIdx.x; i < 16 * 64; i += 128) {
    int rr = i >> 6, cc = i & 63;
    int g = row_base + rr;
    float hv = 0.f, pv = 0.f;
    if (g < n) {
      hv = h[(size_t)g * DD + cc];
      pv = p[(size_t)g * DD + cc];
    }
    ht[rr * LDS_STRIDE + cc] = hv;
    pt[rr * LDS_STRIDE + cc] = pv;
  }
  __syncthreads();
  const int col_base = (threadIdx.x >> 5) * 16;
  v8f acc;

  // A1_h = [h,p] @ A1_w + A1_b   (A1_w is [128,64]: rows 0..63 multiply h, 64..127 multiply p)
  acc = bias_frag(A1b, col_base);
  acc = wmma_k64(ht, A1w, col_base, acc);
  acc = wmma_k64(pt, A1w + 64 * DD, col_base, acc);
  store_tile(A1h, row_base, col_base, acc, n);

  // A2_hp
  acc = bias_frag(A2b, col_base);
  acc = wmma_k64(ht, A2w, col_base, acc);
  acc = wmma_k64(pt, A2w + 64 * DD, col_base, acc);
  store_tile(A2hp, row_base, col_base, acc, n);

  // B1_h = h @ B1_w + b
  acc = bias_frag(B1b, col_base);
  acc = wmma_k64(ht, B1w, col_base, acc);
  store_tile(B1h, row_base, col_base, acc, n);

  // B2_h
  acc = bias_frag(B2b, col_base);
  acc = wmma_k64(ht, B2w, col_base, acc);
  store_tile(B2h, row_base, col_base, acc, n);

  // C1_p = p @ C1_w + b
  acc = bias_frag(C1b, col_base);
  acc = wmma_k64(pt, C1w, col_base, acc);
  store_tile(C1p, row_base, col_base, acc, n);

  // C2_p
  acc = bias_frag(C2b, col_base);
  acc = wmma_k64(pt, C2w, col_base, acc);
  store_tile(C2p, row_base, col_base, acc, n);
}

// ---------------- K2: edge GEMM B3_e = e @ B3_w + b ----------------
__global__ __launch_bounds__(128) void edge_gemm_kernel(
    const float* __restrict__ e, const float* __restrict__ B3w,
    const float* __restrict__ B3b, float* __restrict__ out, int ecount) {
  __shared__ float et[16 * LDS_STRIDE];
  const int row_base = blockIdx.x * 16;
  for (int i = threadIdx.x; i < 16 * 64; i += 128) {
    int rr = i >> 6, cc = i & 63;
    int g = row_base + rr;
    et[rr * LDS_STRIDE + cc] = (g < ecount) ? e[(size_t)g * DD + cc] : 0.f;
  }
  __syncthreads();
  const int col_base = (threadIdx.x >> 5) * 16;
  v8f acc = bias_frag(B3b, col_base);
  acc = wmma_k64(et, B3w, col_base, acc);
  store_tile(out, row_base, col_base, acc, ecount);
}

// ---------------- K3: hat_eta (in place over B3_e) + sigma segment sum ----------------
__global__ __launch_bounds__(256) void edge_pass1_kernel(
    const float* __restrict__ B1h, const float* __restrict__ B2h,
    float* __restrict__ hat, const int* __restrict__ src,
    const int* __restrict__ dst, float* __restrict__ sum_sigma, long long ecount) {
  long long idx = (long long)blockIdx.x * 256 + threadIdx.x;
  long long total = ecount * DD;
  if (idx >= total) return;
  int col = (int)(idx & 63);
  long long eidx = idx >> 6;
  int s = src[eidx], d = dst[eidx];
  float v = B1h[(size_t)s * DD + col] + B2h[(size_t)d * DD + col] + hat[idx];
  hat[idx] = v;
  float sig = 1.f / (1.f + __expf(-v));
  atomicAdd(&sum_sigma[(size_t)d * DD + col], sig);
}

// ---------------- K4: eta-weighted segment sums ----------------
__global__ __launch_bounds__(256) void edge_pass2_kernel(
    const float* __restrict__ hat, const float* __restrict__ A2hp,
    const float* __restrict__ C2p, const int* __restrict__ src,
    const int* __restrict__ dst, const float* __restrict__ sum_sigma,
    float* __restrict__ h_sum, float* __restrict__ p_sum, long long ecount) {
  long long idx = (long long)blockIdx.x * 256 + threadIdx.x;
  long long total = ecount * DD;
  if (idx >= total) return;
  int col = (int)(idx & 63);
  long long eidx = idx >> 6;
  int s = src[eidx], d = dst[eidx];
  float v = hat[idx];
  float sig = 1.f / (1.f + __expf(-v));
  float eta = sig / (sum_sigma[(size_t)d * DD + col] + EPS_ETA);
  atomicAdd(&h_sum[(size_t)d * DD + col], eta * A2hp[(size_t)s * DD + col]);
  atomicAdd(&p_sum[(size_t)d * DD + col], eta * C2p[(size_t)s * DD + col]);
}

// ---------------- K5a: h_new/p_new + h column stats ----------------
__global__ __launch_bounds__(256) void node_combine_stats_kernel(
    const float* __restrict__ A1h, const float* __restrict__ C1p,
    float* __restrict__ h_sum /* becomes h_new */,
    float* __restrict__ p_sum /* becomes p_new */,
    float* __restrict__ stats, int n) {
  __shared__ float s1[4][64];
  __shared__ float s2[4][64];
  const int col = threadIdx.x & 63;
  const int rsub = threadIdx.x >> 6;
  const int row0 = blockIdx.x * 256;
  float sum = 0.f, ssq = 0.f;
  for (int r = rsub; r < 256; r += 4) {
    int row = row0 + r;
    if (row < n) {
      size_t idx = (size_t)row * DD + col;
      float hn = A1h[idx] + h_sum[idx];
      h_sum[idx] = hn;
      float pn = C1p[idx] + p_sum[idx];
      p_sum[idx] = pn;
      sum += hn;
      ssq += hn * hn;
    }
  }
  s1[rsub][col] = sum;
  s2[rsub][col] = ssq;
  __syncthreads();
  if (rsub == 0) {
    atomicAdd(&stats[col], s1[0][col] + s1[1][col] + s1[2][col] + s1[3][col]);
    atomicAdd(&stats[64 + col], s2[0][col] + s2[1][col] + s2[2][col] + s2[3][col]);
  }
}

// ---------------- K5b: e column stats over hat_eta ----------------
__global__ __launch_bounds__(256) void edge_stats_kernel(
    const float* __restrict__ hat, float* __restrict__ stats, int ecount) {
  __shared__ float s1[4][64];
  __shared__ float s2[4][64];
  const int col = threadIdx.x & 63;
  const int rsub = threadIdx.x >> 6;
  const int row0 = blockIdx.x * 1024;
  float sum = 0.f, ssq = 0.f;
  for (int r = rsub; r < 1024; r += 4) {
    int row = row0 + r;
    if (row < ecount) {
      float v = hat[(size_t)row * DD + col];
      sum += v;
      ssq += v * v;
    }
  }
  s1[rsub][col] = sum;
  s2[rsub][col] = ssq;
  __syncthreads();
  if (rsub == 0) {
    atomicAdd(&stats[128 + col], s1[0][col] + s1[1][col] + s1[2][col] + s1[3][col]);
    atomicAdd(&stats[192 + col], s2[0][col] + s2[1][col] + s2[2][col] + s2[3][col]);
  }
}

// ---------------- K5c: finalize mu / rstd ----------------
__global__ void finalize_stats_kernel(float* stats, int n, long long ecount) {
  int c = threadIdx.x;  // 64 threads
  if (c >= 64) return;
  float invn = 1.f / (float)n;
  float muh = stats[c] * invn;
  float varh = stats[64 + c] * invn - muh * muh;
  stats[256 + c] = muh;
  stats[320 + c] = rsqrtf(varh + EPS_BN);
  float inve = 1.f / (float)ecount;
  float mue = stats[128 + c] * inve;
  float vare = stats[192 + c] * inve - mue * mue;
  stats[384 + c] = mue;
  stats[448 + c] = rsqrtf(vare + EPS_BN);
}

// ---------------- K6a: node outputs ----------------
__global__ __launch_bounds__(256) void node_out_kernel(
    const float* __restrict__ h_in, const float* __restrict__ p_in,
    const float* __restrict__ hnew, const float* __restrict__ pnew,
    const float* __restrict__ stats, const float* __restrict__ gamma,
    const float* __restrict__ beta, float* __restrict__ out_h,
    float* __restrict__ out_p, long long nd) {
  long long i = (long long)blockIdx.x * 256 + threadIdx.x;
  if (i >= nd) return;
  int col = (int)(i & 63);
  float hn = hnew[i];
  float v = (hn - stats[256 + col]) * stats[320 + col] * gamma[col] + beta[col];
  out_h[i] = h_in[i] + fmaxf(v, 0.f);
  out_p[i] = p_in[i] + tanhf(pnew[i]);
}

// ---------------- K6b: edge outputs ----------------
__global__ __launch_bounds__(256) void edge_out_kernel(
    const float* __restrict__ e_in, const float* __restrict__ hat,
    const float* __restrict__ stats, const float* __restrict__ gamma,
    const float* __restrict__ beta, float* __restrict__ out_e, long long ed) {
  long long i = (long long)blockIdx.x * 256 + threadIdx.x;
  if (i >= ed) return;
  int col = (int)(i & 63);
  float v = (hat[i] - stats[384 + col]) * stats[448 + col] * gamma[col] + beta[col];
  out_e[i] = e_in[i] + fmaxf(v, 0.f);
}

extern "C" void kernel_launch(void* const* d_in, const int* in_sizes, int n_in,
                              void* d_out, int out_size, void* d_ws, size_t ws_size,
                              hipStream_t stream) {
  const float* h = (const float*)d_in[0];
  const float* p = (const float*)d_in[1];
  const float* e = (const float*)d_in[2];
  // d_in[3] = snorm_n (unused by the reference)
  const int* src = (const int*)d_in[4];
  const int* dst = (const int*)d_in[5];
  const float* A1w = (const float*)d_in[6];
  const float* A1b = (const float*)d_in[7];
  const float* A2w = (const float*)d_in[8];
  const float* A2b = (const float*)d_in[9];
  const float* B1w = (const float*)d_in[10];
  const float* B1b = (const float*)d_in[11];
  const float* B2w = (const float*)d_in[12];
  const float* B2b = (const float*)d_in[13];
  const float* B3w = (const float*)d_in[14];
  const float* B3b = (const float*)d_in[15];
  const float* C1w = (const float*)d_in[16];
  const float* C1b = (const float*)d_in[17];
  const float* C2w = (const float*)d_in[18];
  const float* C2b = (const float*)d_in[19];
  const float* bn_h_gamma = (const float*)d_in[20];
  const float* bn_h_beta = (const float*)d_in[21];
  const float* bn_e_gamma = (const float*)d_in[22];
  const float* bn_e_beta = (const float*)d_in[23];

  const int N = in_sizes[0] / DD;
  const int E = in_sizes[2] / DD;
  const size_t nd = (size_t)N * DD;
  const size_t ed = (size_t)E * DD;

  float* ws = (float*)d_ws;
  float* A1h = ws;
  float* A2hp = A1h + nd;
  float* B1h = A2hp + nd;
  float* B2h = B1h + nd;
  float* C1p = B2h + nd;
  float* C2p = C1p + nd;
  float* hat = C2p + nd;        // E*D; holds B3_e then hat_eta in place
  float* sum_sigma = hat + ed;  // zeroed
  float* h_sum = sum_sigma + nd;  // zeroed; becomes h_new
  float* p_sum = h_sum + nd;      // zeroed; becomes p_new
  float* stats = p_sum + nd;      // 512 floats, zeroed

  // zero sum_sigma .. stats end (contiguous)
  zero_kernel<<<2048, 256, 0, stream>>>(sum_sigma, 3 * nd + 512);

  node_linear_kernel<<<(N + 15) / 16, 128, 0, stream>>>(
      h, p, A1w, A1b, A2w, A2b, B1w, B1b, B2w, B2b, C1w, C1b, C2w, C2b,
      A1h, A2hp, B1h, B2h, C1p, C2p, N);

  edge_gemm_kernel<<<(E + 15) / 16, 128, 0, stream>>>(e, B3w, B3b, hat, E);

  long long total_e = (long long)E * DD;
  int ge = (int)((total_e + 255) / 256);
  edge_pass1_kernel<<<ge, 256, 0, stream>>>(B1h, B2h, hat, src, dst, sum_sigma, E);
  edge_pass2_kernel<<<ge, 256, 0, stream>>>(hat, A2hp, C2p, src, dst, sum_sigma,
                                            h_sum, p_sum, E);

  node_combine_stats_kernel<<<(N + 255) / 256, 256, 0, stream>>>(A1h, C1p, h_sum,
                                                                 p_sum, stats, N);
  edge_stats_kernel<<<(E + 1023) / 1024, 256, 0, stream>>>(hat, stats, E);
  finalize_stats_kernel<<<1, 64, 0, stream>>>(stats, N, (long long)E);

  float* out_h = (float*)d_out;
  float* out_p = out_h + nd;
  float* out_e = out_p + nd;
  long long total_n = (long long)N * DD;
  int gn = (int)((total_n + 255) / 256);
  node_out_kernel<<<gn, 256, 0, stream>>>(h, p, h_sum, p_sum, stats, bn_h_gamma,
                                          bn_h_beta, out_h, out_p, total_n);
  edge_out_kernel<<<ge, 256, 0, stream>>>(e, hat, stats, bn_e_gamma, bn_e_beta,
                                          out_e, total_e);
}